// LatentEncoderNTokensSpectralFilter_23802708754943
// MI455X (gfx1250) — compile-verified
//
#include <hip/hip_runtime.h>
#include <math.h>

// ---------------------------------------------------------------------------
// Reference collapses algebraically: M = C^{-1} and the band mask depends only
// on (l, sector), so IDCT(mask*DCT(x)) = mask*x.  Remaining work:
//   y[b,l,n,o] = tanh( mask(l,n) * <enc[b,l,n,:], W[o,:]> + bias[o] ),  l < 32
// => fp32 GEMM  M=4096, N=256, K=256  via V_WMMA_F32_16X16X4_F32.
// This version: 32x32 output tile per wave (2x2 register blocking) so the
// inner loop is 4 global_load_b64 : 4 WMMA  (1 load per WMMA, was 2).
// ---------------------------------------------------------------------------

typedef __attribute__((ext_vector_type(2))) float v2f;
typedef __attribute__((ext_vector_type(8))) float v8f;

#define L_DIM   512
#define D_DIM   1024
#define SSIZE   256
#define LATENT  256
#define COEF    0.1f

// Second row-block (m+16) => same (bb,sec), l+4  => +4*D_DIM floats = 16384 B.
// Second col-block (n+16) => +16*SSIZE floats            = 16384 B.
#define BLK_OFF_FLOATS (4 * D_DIM)     // == 16 * SSIZE == 4096 floats

__device__ __forceinline__ float band_scale(int rowM)
{
    const int rem = rowM & 127;        // 128 rows per batch image
    const int l   = rem >> 2;
    const int sec = rem & 3;
    // bands restricted to l < 32:
    //   sector 0 (0,64)->1.0 ; sector 1 (16,128)->1.0 iff l>=16 ; 2,3 -> 0.1
    return (sec == 0) ? 1.0f : ((sec == 1) && (l >= 16)) ? 1.0f : COEF;
}

__global__ __launch_bounds__(128)
void spectral_gemm_wmma(const float* __restrict__ enc,
                        const float* __restrict__ W,
                        const float* __restrict__ bias,
                        float* __restrict__ out)
{
    const int lane = threadIdx.x & 31;
    const int wave = threadIdx.x >> 5;
    const int tile = blockIdx.x * 4 + wave;     // 1024 tiles of 32x32
    const int nt2  = tile & 7;                  // 8 tiles along N (=256)
    const int mt2  = tile >> 3;                 // 128 tiles along M (=4096)

    const int half = lane >> 4;                 // 0: lanes 0-15, 1: lanes 16-31
    const int ln   = lane & 15;
    const int kh   = half * 2;                  // K offset within a K=4 step

    // ---- A base: row = mt2*32 + ln ; row = ((bb*32 + l)*4 + sec)
    const int rowA = mt2 * 32 + ln;
    const int bb   = rowA >> 7;
    const int rem  = rowA & 127;
    const int l0   = rem >> 2;
    const int sec  = rem & 3;
    const float* __restrict__ aptr =
        enc + ((size_t)bb * (size_t)(L_DIM * D_DIM)
             + (size_t)l0 * (size_t)D_DIM
             + (size_t)sec * (size_t)SSIZE)
            + kh;

    // ---- B base: col = nt2*32 + ln ; B[k,n] = W[col,k]
    const int col = nt2 * 32 + ln;
    const float* __restrict__ bptr = W + (size_t)col * (size_t)SSIZE + kh;

    // ---- K loop: fragments per ISA 7.12.2 (32-bit A 16x4 / B 4x16):
    // lanes 0-15 carry K{0,1}, lanes 16-31 carry K{2,3} -> one b64 per frag.
    v8f c00 = {}, c01 = {}, c10 = {}, c11 = {};
    #pragma unroll 4
    for (int k0 = 0; k0 < SSIZE; k0 += 4) {
        v2f a0 = *(const v2f*)(aptr + k0);                   // rows m..m+15
        v2f a1 = *(const v2f*)(aptr + k0 + BLK_OFF_FLOATS);  // rows m+16..m+31
        v2f b0 = *(const v2f*)(bptr + k0);                   // cols n..n+15
        v2f b1 = *(const v2f*)(bptr + k0 + BLK_OFF_FLOATS);  // cols n+16..n+31
        c00 = __builtin_amdgcn_wmma_f32_16x16x4_f32(false, a0, false, b0,
                                                    (short)0, c00, false, false);
        c01 = __builtin_amdgcn_wmma_f32_16x16x4_f32(false, a0, false, b1,
                                                    (short)0, c01, false, false);
        c10 = __builtin_amdgcn_wmma_f32_16x16x4_f32(false, a1, false, b0,
                                                    (short)0, c10, false, false);
        c11 = __builtin_amdgcn_wmma_f32_16x16x4_f32(false, a1, false, b1,
                                                    (short)0, c11, false, false);
    }

    const float bc0 = bias[col];
    const float bc1 = bias[col + 16];

    // ---- Epilogue: C/D layout -> VGPR v holds (M = v + 8*half, N = ln).
    const int mBase = mt2 * 32 + 8 * half;
    #pragma unroll
    for (int v = 0; v < 8; ++v) {
        const int r0 = mBase + v;          // rows of c00/c01
        const int r1 = r0 + 16;            // rows of c10/c11
        const float s0 = band_scale(r0);
        const float s1 = band_scale(r1);
        out[(size_t)r0 * LATENT + col]      = tanhf(s0 * c00[v] + bc0);
        out[(size_t)r0 * LATENT + col + 16] = tanhf(s0 * c01[v] + bc1);
        out[(size_t)r1 * LATENT + col]      = tanhf(s1 * c10[v] + bc0);
        out[(size_t)r1 * LATENT + col + 16] = tanhf(s1 * c11[v] + bc1);
    }
}

extern "C" void kernel_launch(void* const* d_in, const int* in_sizes, int n_in,
                              void* d_out, int out_size, void* d_ws, size_t ws_size,
                              hipStream_t stream)
{
    const float* enc  = (const float*)d_in[0];   // [32, 512, 1024] fp32
    const float* W    = (const float*)d_in[1];   // [256, 256] fp32
    const float* bias = (const float*)d_in[2];   // [256] fp32
    float* out        = (float*)d_out;           // [32, 32, 4, 256] fp32

    // 1024 wave-tiles (32x32), 4 waves (128 threads) per block -> 256 blocks
    dim3 grid(256);
    dim3 block(128);
    hipLaunchKernelGGL(spectral_gemm_wmma, grid, block, 0, stream,
                       enc, W, bias, out);
}